// NodeLayer_70884140253682
// MI455X (gfx1250) — compile-verified
//
#include <hip/hip_runtime.h>

// ---------------------------------------------------------------------------
// GraphNet NodeLayer for MI455X (gfx1250, wave32, WMMA).
//
// Pipeline (BN folded into the following Linear once batch stats are known):
//   [stats of edge input] -> fold W1 -> GEMM1(relu,+stats) -> fold W2 ->
//   GEMM2(relu,+stats, in-place) -> fold W3 -> GEMM3(scatter-add to nodes)
//   -> build node input (x | mean-agg) + stats -> fold -> GEMM(96->64,relu,+stats)
//   -> fold -> GEMM(64->64,relu,+stats,in-place) -> fold -> GEMM(64->64) -> out
//
// GEMMs use V_WMMA_F32_16X16X4_F32; one wave = 16x64 output tile.
// B kept in a K-paired LDS layout (pair stride 160 floats) so each fragment is
// a single conflict-free ds_load_b64; A staged with row stride KDIM+4 so
// fragment loads spread over all 64 LDS banks.
// ---------------------------------------------------------------------------

#define NE_C 1600000
#define NN_C 100000
#define BPAIR 160  // floats per K-pair row of B in LDS/ws (128 used + pad)

typedef float v2f __attribute__((ext_vector_type(2)));
typedef float v4f __attribute__((ext_vector_type(4)));
typedef float v8f __attribute__((ext_vector_type(8)));

// ---------------------------------------------------------------------------
// Pass 0: per-column sum / sumsq of the edge-MLP input concat(x[row], e_attr),
// plus in-degree counts of target nodes (for scatter-mean).
// ---------------------------------------------------------------------------
__global__ __launch_bounds__(256) void edge_input_stats_kernel(
    const float* __restrict__ X, const float* __restrict__ EA,
    const int* __restrict__ EIDX, float* __restrict__ gsum,
    float* __restrict__ gsq, float* __restrict__ cnt, int NE)
{
  __shared__ float sSum[64], sSq[64];
  const int tid = threadIdx.x;
  if (tid < 64) { sSum[tid] = 0.f; sSq[tid] = 0.f; }
  __syncthreads();
  const int c = tid & 63, sub = tid >> 6;
  float ls = 0.f, lq = 0.f;
  for (long e = (long)blockIdx.x * 4 + sub; e < NE; e += (long)gridDim.x * 4) {
    float v = (c < 32) ? X[(size_t)EIDX[e] * 32 + c]
                       : EA[(size_t)e * 32 + (c - 32)];
    ls += v; lq += v * v;
    if (c == 0) atomicAdd(&cnt[EIDX[NE + e]], 1.0f);
  }
  atomicAdd(&sSum[c], ls);
  atomicAdd(&sSq[c], lq);
  __syncthreads();
  if (tid < 64) { atomicAdd(&gsum[tid], sSum[tid]); atomicAdd(&gsq[tid], sSq[tid]); }
}

// ---------------------------------------------------------------------------
// Fold BN(mu,var,g,b_bn) + Linear(W,b) into the K-paired B layout:
//   Bout[(k>>1)*BPAIR + n*2 + (k&1)] = W[n][k] * s[k]*g[k]
// and bout[n] = b[n] + sum_k (b_bn[k] - mu[k]*s[k]*g[k]) * W[n][k].
// ---------------------------------------------------------------------------
__global__ __launch_bounds__(128) void fold_kernel(
    const float* __restrict__ g, const float* __restrict__ bbn,
    const float* __restrict__ W, const float* __restrict__ b,
    const float* __restrict__ gsum, const float* __restrict__ gsq,
    float invN, int K, float* __restrict__ Bout, float* __restrict__ bout)
{
  __shared__ float scale[96], shift[96];
  const int tid = threadIdx.x;
  if (tid < K) {
    float mu  = gsum[tid] * invN;
    float var = gsq[tid] * invN - mu * mu;
    float s   = rsqrtf(var + 1e-5f) * g[tid];
    scale[tid] = s;
    shift[tid] = bbn[tid] - mu * s;
  }
  __syncthreads();
  for (int i = tid; i < K * 64; i += 128) {
    int k = i >> 6, n = i & 63;
    Bout[(k >> 1) * BPAIR + n * 2 + (k & 1)] = W[n * K + k] * scale[k];
  }
  if (tid < 64) {
    float acc = b[tid];
    for (int k = 0; k < K; ++k) acc += shift[k] * W[tid * K + k];
    bout[tid] = acc;
  }
}

// ---------------------------------------------------------------------------
// Build node-MLP input: in2[n] = concat(x[n], summed[n]/max(cnt,1)); 96 cols,
// accumulate its column stats. blockDim = 96 (3 waves), thread = fixed column.
// ---------------------------------------------------------------------------
__global__ __launch_bounds__(96) void node_input_kernel(
    const float* __restrict__ X, const float* __restrict__ summed,
    const float* __restrict__ cnt, float* __restrict__ in2,
    float* __restrict__ gsum, float* __restrict__ gsq, int NN)
{
  const int c = threadIdx.x;
  float ls = 0.f, lq = 0.f;
  for (int n = blockIdx.x; n < NN; n += gridDim.x) {
    float invc = 1.0f / fmaxf(cnt[n], 1.0f);
    float v = (c < 32) ? X[(size_t)n * 32 + c]
                       : summed[(size_t)n * 64 + (c - 32)] * invc;
    in2[(size_t)n * 96 + c] = v;
    ls += v; lq += v * v;
  }
  atomicAdd(&gsum[c], ls);
  atomicAdd(&gsq[c], lq);
}

// ---------------------------------------------------------------------------
// WMMA GEMM: out[M x 64] = act(A[M x KDIM] @ B[KDIM x 64] + bias)
//  GATHER : A row m = concat(x[EIDX[m]], edge_attr[m])       (edge layer 1)
//  SCATTER: atomically add output row m into out[EIDX[NE+m]] (edge layer 3)
//  STATS  : accumulate column sum/sumsq of the output (for the next BN)
// Block = 256 threads = 8 waves; each wave owns one 16-row M tile x N=64.
// Fragment layouts per CDNA5 ISA 7.12.2 (f32 A 16x4 / B 4x16 / C 16x16).
// ---------------------------------------------------------------------------
template <int KDIM, bool GATHER, bool RELU, bool STATS, bool SCATTER>
__global__ __launch_bounds__(256) void gemm_kernel(
    const float* __restrict__ A, const float* __restrict__ X,
    const float* __restrict__ EA, const int* __restrict__ EIDX,
    const float* __restrict__ Bm, const float* __restrict__ bias,
    float* __restrict__ out, float* __restrict__ gsum, float* __restrict__ gsq,
    int numTiles, int NE)
{
  constexpr int ASTR = KDIM + 4;          // padded A row stride (bank spread)
  constexpr int BSZ  = (KDIM / 2) * BPAIR;
  __shared__ float sB[BSZ];
  __shared__ float sBias[64];
  __shared__ float sA[8][16 * ASTR];
  __shared__ float sSum[64], sSq[64];
  __shared__ int   sTgt[8][16];

  const int tid = threadIdx.x;
  for (int i = tid; i < BSZ; i += 256) sB[i] = Bm[i];
  if (tid < 64) {
    sBias[tid] = bias[tid];
    if (STATS) { sSum[tid] = 0.f; sSq[tid] = 0.f; }
  }

  const int wave = tid >> 5;
  const int lane = tid & 31;
  const int t = blockIdx.x * 8 + wave;
  const bool active = t < numTiles;

  if (active) {
    const int er = lane >> 1, half = lane & 1;
    const int m = t * 16 + er;
    if constexpr (GATHER) {
      if (half == 0) {
        const v4f* src = (const v4f*)(X + (size_t)EIDX[m] * 32);
        v4f* dst = (v4f*)(&sA[wave][er * ASTR]);
#pragma unroll
        for (int i = 0; i < 8; ++i) dst[i] = src[i];
      } else {
        const v4f* src = (const v4f*)(EA + (size_t)m * 32);
        v4f* dst = (v4f*)(&sA[wave][er * ASTR + 32]);
#pragma unroll
        for (int i = 0; i < 8; ++i) dst[i] = src[i];
      }
    } else {
      constexpr int HK = KDIM / 2;  // 32 or 48 floats per half-row
      const v4f* src = (const v4f*)(A + (size_t)m * KDIM + half * HK);
      v4f* dst = (v4f*)(&sA[wave][er * ASTR + half * HK]);
#pragma unroll
      for (int i = 0; i < HK / 4; ++i) dst[i] = src[i];
    }
    if constexpr (SCATTER) {
      if (lane < 16) sTgt[wave][lane] = EIDX[NE + t * 16 + lane];
    }
  }
  __syncthreads();

  if (active) {
    const int l15 = lane & 15;
    const int h   = lane >> 4;  // half-wave selects K pair / M offset
    const v8f z = {0.f, 0.f, 0.f, 0.f, 0.f, 0.f, 0.f, 0.f};
    v8f acc[4] = {z, z, z, z};

    for (int k = 0; k < KDIM; k += 4) {
      // A 16x4 f32: lane(0-15)=row M, VGPR0/1 = K{0,1}; lanes 16-31 = K{2,3}
      v2f a = *(const v2f*)(&sA[wave][l15 * ASTR + k + 2 * h]);
      // K-pair (k+2h)/2 holds {B[k+2h][n], B[k+1+2h][n]} adjacently.
      const float* bp = &sB[((k >> 1) + h) * BPAIR + l15 * 2];
#pragma unroll
      for (int nt = 0; nt < 4; ++nt) {
        v2f b = *(const v2f*)(bp + nt * 32);
        acc[nt] = __builtin_amdgcn_wmma_f32_16x16x4_f32(
            false, a, false, b, (short)0, acc[nt], false, false);
      }
    }

#pragma unroll
    for (int nt = 0; nt < 4; ++nt) {
      const int col = nt * 16 + l15;
      const float bv = sBias[col];
      float ls = 0.f, lq = 0.f;
#pragma unroll
      for (int v = 0; v < 8; ++v) {
        float val = acc[nt][v] + bv;
        if (RELU) val = fmaxf(val, 0.f);
        const int m = t * 16 + v + 8 * h;  // C/D: M = vgpr + 8*half
        if constexpr (SCATTER) {
          atomicAdd(&out[(size_t)sTgt[wave][v + 8 * h] * 64 + col], val);
        } else {
          out[(size_t)m * 64 + col] = val;
        }
        if (STATS) { ls += val; lq += val * val; }
      }
      if (STATS) { atomicAdd(&sSum[col], ls); atomicAdd(&sSq[col], lq); }
    }
  }

  if constexpr (STATS) {
    __syncthreads();
    if (tid < 64) { atomicAdd(&gsum[tid], sSum[tid]); atomicAdd(&gsq[tid], sSq[tid]); }
  }
}

// ---------------------------------------------------------------------------
extern "C" void kernel_launch(void* const* d_in, const int* in_sizes, int n_in,
                              void* d_out, int out_size, void* d_ws, size_t ws_size,
                              hipStream_t stream)
{
  (void)in_sizes; (void)n_in; (void)out_size; (void)ws_size;
  const float* x  = (const float*)d_in[0];
  const int*   ei = (const int*)  d_in[1];   // [2, NE] (harness passes int32)
  const float* ea = (const float*)d_in[2];
  // d_in[3]=u, d_in[4]=batch : unused by the reference computation.
  // mlp1[0..2], mlp2[0..2], each {bn_g, bn_b, W, b} in setup_inputs order.
  const float* P[6][4];
  for (int i = 0; i < 6; ++i)
    for (int j = 0; j < 4; ++j)
      P[i][j] = (const float*)d_in[5 + 4 * i + j];

  char* base = (char*)d_ws;
  size_t off = 0;
  auto take = [&](size_t bytes) -> void* {
    void* p = base + off;
    off = (off + bytes + 255) & ~(size_t)255;
    return p;
  };
  const int wbSlot = 48 * BPAIR + 64;  // padded B (max K=96) + bias
  float* hbuf   = (float*)take((size_t)NE_C * 64 * sizeof(float)); // 409.6 MB
  float* summed = (float*)take((size_t)NN_C * 64 * sizeof(float)); // 25.6 MB
  float* cntf   = (float*)take((size_t)NN_C * sizeof(float));
  float* in2    = (float*)take((size_t)NN_C * 96 * sizeof(float)); // 38.4 MB
  float* nh     = (float*)take((size_t)NN_C * 64 * sizeof(float));
  float* stats  = (float*)take(6 * 192 * sizeof(float));
  float* wb     = (float*)take(6 * (size_t)wbSlot * sizeof(float));

  auto ssum = [&](int i) { return stats + i * 192; };
  auto ssq  = [&](int i) { return stats + i * 192 + 96; };
  auto wB   = [&](int i) { return wb + (size_t)i * wbSlot; };
  auto wbb  = [&](int i) { return wb + (size_t)i * wbSlot + 48 * BPAIR; };

  hipMemsetAsync(summed, 0, (size_t)NN_C * 64 * sizeof(float), stream);
  hipMemsetAsync(cntf,   0, (size_t)NN_C * sizeof(float), stream);
  hipMemsetAsync(stats,  0, 6 * 192 * sizeof(float), stream);

  const int eTiles = NE_C / 16;           // 100000
  const int eGrid  = eTiles / 8;          // 12500
  const int nTiles = NN_C / 16;           // 6250
  const int nGrid  = (nTiles + 7) / 8;    // 782

  // ---- edge MLP ----
  edge_input_stats_kernel<<<2048, 256, 0, stream>>>(x, ea, ei, ssum(0), ssq(0), cntf, NE_C);
  fold_kernel<<<1, 128, 0, stream>>>(P[0][0], P[0][1], P[0][2], P[0][3],
                                     ssum(0), ssq(0), 1.0f / NE_C, 64, wB(0), wbb(0));
  gemm_kernel<64, true, true, true, false><<<eGrid, 256, 0, stream>>>(
      nullptr, x, ea, ei, wB(0), wbb(0), hbuf, ssum(1), ssq(1), eTiles, NE_C);
  fold_kernel<<<1, 128, 0, stream>>>(P[1][0], P[1][1], P[1][2], P[1][3],
                                     ssum(1), ssq(1), 1.0f / NE_C, 64, wB(1), wbb(1));
  gemm_kernel<64, false, true, true, false><<<eGrid, 256, 0, stream>>>(
      hbuf, nullptr, nullptr, nullptr, wB(1), wbb(1), hbuf, ssum(2), ssq(2), eTiles, NE_C);
  fold_kernel<<<1, 128, 0, stream>>>(P[2][0], P[2][1], P[2][2], P[2][3],
                                     ssum(2), ssq(2), 1.0f / NE_C, 64, wB(2), wbb(2));
  gemm_kernel<64, false, false, false, true><<<eGrid, 256, 0, stream>>>(
      hbuf, nullptr, nullptr, ei, wB(2), wbb(2), summed, nullptr, nullptr, eTiles, NE_C);

  // ---- scatter-mean + node MLP ----
  node_input_kernel<<<1024, 96, 0, stream>>>(x, summed, cntf, in2, ssum(3), ssq(3), NN_C);
  fold_kernel<<<1, 128, 0, stream>>>(P[3][0], P[3][1], P[3][2], P[3][3],
                                     ssum(3), ssq(3), 1.0f / NN_C, 96, wB(3), wbb(3));
  gemm_kernel<96, false, true, true, false><<<nGrid, 256, 0, stream>>>(
      in2, nullptr, nullptr, nullptr, wB(3), wbb(3), nh, ssum(4), ssq(4), nTiles, NE_C);
  fold_kernel<<<1, 128, 0, stream>>>(P[4][0], P[4][1], P[4][2], P[4][3],
                                     ssum(4), ssq(4), 1.0f / NN_C, 64, wB(4), wbb(4));
  gemm_kernel<64, false, true, true, false><<<nGrid, 256, 0, stream>>>(
      nh, nullptr, nullptr, nullptr, wB(4), wbb(4), nh, ssum(5), ssq(5), nTiles, NE_C);
  fold_kernel<<<1, 128, 0, stream>>>(P[5][0], P[5][1], P[5][2], P[5][3],
                                     ssum(5), ssq(5), 1.0f / NN_C, 64, wB(5), wbb(5));
  gemm_kernel<64, false, false, false, false><<<nGrid, 256, 0, stream>>>(
      nh, nullptr, nullptr, nullptr, wB(5), wbb(5), (float*)d_out, nullptr, nullptr, nTiles, NE_C);
}